// DGRec_22445499088956
// MI455X (gfx1250) — compile-verified
//
#include <hip/hip_runtime.h>
#include <math.h>

// DGRec: DSP(FFT low-pass + LN) + batch attention + GGNN + GAT, all fp32.
// Matrix ops routed through V_WMMA_F32_16X16X4_F32 (full fp32 precision).

typedef __attribute__((ext_vector_type(2))) float v2f;
typedef __attribute__((ext_vector_type(8))) float v8f;

#define B_  1024
#define L_  20
#define D_  50
#define BL_ (B_ * L_)   // 20480
#define M_  5120

static __device__ __forceinline__ v8f wmma4(v2f a, v2f b, v8f c) {
  // D(16x16,f32) = A(16x4,f32) * B(4x16,f32) + C
  return __builtin_amdgcn_wmma_f32_16x16x4_f32(false, a, false, b, (short)0, c,
                                               false, false);
}

// ---------------------------------------------------------------------------
// K1: emb = item_emb[session] + pop_emb[pop];  h = item_emb[node_items]
// ---------------------------------------------------------------------------
__global__ __launch_bounds__(256) void gather_kernel(
    const float* __restrict__ item_emb, const float* __restrict__ pop_emb,
    const int* __restrict__ sess, const int* __restrict__ pop,
    const int* __restrict__ nodes, float* __restrict__ emb,
    float* __restrict__ h) {
  int i = blockIdx.x * 256 + threadIdx.x;
  if (i >= BL_ * D_) return;
  int r = i / D_, d = i - r * D_;
  emb[i] = item_emb[sess[r] * D_ + d] + pop_emb[pop[r] * D_ + d];
  h[i]   = item_emb[nodes[r] * D_ + d];
}

// ---------------------------------------------------------------------------
// K2: DSP branch. rfft/irfft with freqs >= 2 zeroed == fixed 20x20 kernel:
//     low[t] = sum_u (1 + 2cos(2pi(t-u)/20))/20 * emb[u]
// then x = low + beta^2*(emb-low) + emb ; layernorm ; hidden = ALPHA*dsp
// ---------------------------------------------------------------------------
__global__ __launch_bounds__(256) void dsp_kernel(
    const float* __restrict__ emb, const float* __restrict__ beta,
    const float* __restrict__ ln_w, const float* __restrict__ ln_b,
    float* __restrict__ hidden) {
  int b = blockIdx.x;
  __shared__ float E[L_][D_];
  __shared__ float Lo[L_][D_];
  int tid = threadIdx.x;
  for (int i = tid; i < L_ * D_; i += 256)
    E[i / D_][i % D_] = emb[(size_t)b * (L_ * D_) + i];
  __syncthreads();
  const float W0 = 6.283185307179586f / 20.0f;
  for (int i = tid; i < L_ * D_; i += 256) {
    int t = i / D_, d = i - (i / D_) * D_;
    float s = 0.f;
    #pragma unroll
    for (int u = 0; u < L_; ++u) {
      float kv = (1.0f + 2.0f * __cosf(W0 * (float)(t - u))) * 0.05f;
      s += kv * E[u][d];
    }
    Lo[t][d] = s;
  }
  __syncthreads();
  if (tid < L_) {
    int t = tid;
    float mean = 0.f;
    for (int d = 0; d < D_; ++d) {
      float bb = beta[d];
      float e = E[t][d], lo = Lo[t][d];
      mean += lo + bb * bb * (e - lo) + e;
    }
    mean *= 0.02f;
    float var = 0.f;
    for (int d = 0; d < D_; ++d) {
      float bb = beta[d];
      float e = E[t][d], lo = Lo[t][d];
      float v = lo + bb * bb * (e - lo) + e - mean;
      var += v * v;
    }
    var *= 0.02f;
    float inv = 1.0f / sqrtf(var + 1e-12f);
    for (int d = 0; d < D_; ++d) {
      float bb = beta[d];
      float e = E[t][d], lo = Lo[t][d];
      float v = lo + bb * bb * (e - lo) + e;
      hidden[((size_t)b * L_ + t) * D_ + d] =
          0.9f * (ln_w[d] * (v - mean) * inv + ln_b[d]);
    }
  }
}

// ---------------------------------------------------------------------------
// Generic WMMA row-GEMM: Y[r][e](ldy) = epi( sum_d X[r][d](ldx) * W[e*K+d] + b[e] )
// M multiple of 64 (block = 4 waves x 16 rows). K <= 100 (padded to x4).
// mode 0: store   mode 1: Y += wscale*v   mode 2: relu store
// ---------------------------------------------------------------------------
#define GKP 104
__global__ __launch_bounds__(128) void gemm_wmma(
    const float* __restrict__ X, int ldx, const float* __restrict__ W,
    const float* __restrict__ bias, float* __restrict__ Y, int ldy, int K,
    int N, int mode, float wscale) {
  __shared__ float Xs[64][GKP];   // 26.6 KB
  __shared__ float Wc[GKP][16];   //  6.6 KB
  const int tid = threadIdx.x;
  const int Kp = (K + 3) & ~3;
  const int ntiles = (N + 15) >> 4;
  const int r0 = blockIdx.x * 64;
  for (int i = tid; i < 64 * Kp; i += 128) {
    int r = i / Kp, k = i - r * Kp;
    Xs[r][k] = (k < K) ? X[(size_t)(r0 + r) * ldx + k] : 0.0f;
  }
  const int lane = tid & 31;
  const int wave = tid >> 5;
  const int hi = lane >> 4;     // 0|1: which K half / which M half of C
  const int lo = lane & 15;
  const int rbase = wave * 16;
  const int ksteps = Kp >> 2;
  for (int nt = 0; nt < ntiles; ++nt) {
    __syncthreads();
    for (int i = tid; i < Kp * 16; i += 128) {
      int k = i >> 4, e = i & 15;
      int col = nt * 16 + e;
      Wc[k][e] = (k < K && col < N) ? W[(size_t)col * K + k] : 0.0f;
    }
    __syncthreads();
    v8f acc = {};
    for (int kk = 0; kk < ksteps; ++kk) {
      int kb = (kk << 2) + hi * 2;
      v2f a, b;
      a.x = Xs[rbase + lo][kb];
      a.y = Xs[rbase + lo][kb + 1];
      b.x = Wc[kb][lo];
      b.y = Wc[kb + 1][lo];
      acc = wmma4(a, b, acc);
    }
    int col = nt * 16 + lo;
    if (col < N) {
      float bv = bias ? bias[col] : 0.0f;
      #pragma unroll
      for (int i = 0; i < 8; ++i) {
        int row = r0 + rbase + i + hi * 8;   // C layout: M = i + 8*hi
        float v = acc[i] + bv;
        size_t off = (size_t)row * ldy + col;
        if (mode == 0)      Y[off] = v;
        else if (mode == 1) Y[off] += wscale * v;
        else                Y[off] = fmaxf(v, 0.0f);
      }
    }
  }
}

// ---------------------------------------------------------------------------
// K4: flash attention per (l, 16-row s-tile). qkv rows = (b*20+l)*150.
// scores(l) = (Q*scale) K^T over batch; softmax over t; ctx = P V.
// One wave per block; lanes 0..15 own per-row softmax state.
// ---------------------------------------------------------------------------
__global__ __launch_bounds__(32) void attn_kernel(
    const float* __restrict__ qkv, float* __restrict__ ctx) {
  const int l = blockIdx.y;
  const int s0 = blockIdx.x * 16;
  const int lane = threadIdx.x;
  const int hi = lane >> 4, lo = lane & 15;
  __shared__ float Qs[16][52];
  __shared__ float Ks[16][52];
  __shared__ float Vs[16][64];
  __shared__ float Sl[16][16];
  __shared__ float salpha[16];
  __shared__ float slsum[16];
  const float scale = 0.1414213562373095f;  // 1/sqrt(50)

  for (int i = lane; i < 16 * 52; i += 32) {
    int r = i / 52, d = i - r * 52;
    Qs[r][d] = (d < D_) ? qkv[((size_t)(s0 + r) * L_ + l) * 150 + d] * scale : 0.f;
  }
  v8f O[4];
  #pragma unroll
  for (int n = 0; n < 4; ++n) O[n] = (v8f){};
  float m_i = -3.0e38f, l_i = 0.f;  // valid in lanes 0..15

  for (int tt = 0; tt < B_ / 16; ++tt) {
    const int t0 = tt * 16;
    for (int i = lane; i < 16 * 52; i += 32) {
      int r = i / 52, d = i - r * 52;
      Ks[r][d] = (d < D_) ? qkv[((size_t)(t0 + r) * L_ + l) * 150 + 50 + d] : 0.f;
    }
    for (int i = lane; i < 16 * 64; i += 32) {
      int r = i >> 6, d = i & 63;
      Vs[r][d] = (d < D_) ? qkv[((size_t)(t0 + r) * L_ + l) * 150 + 100 + d] : 0.f;
    }
    // S = Q K^T   (A = Q tile, B[d][t] = Ks[t][d])
    v8f c = (v8f){};
    for (int kk = 0; kk < 13; ++kk) {
      int kb = (kk << 2) + hi * 2;
      v2f a, b;
      a.x = Qs[lo][kb]; a.y = Qs[lo][kb + 1];
      b.x = Ks[lo][kb]; b.y = Ks[lo][kb + 1];
      c = wmma4(a, b, c);
    }
    #pragma unroll
    for (int i = 0; i < 8; ++i) Sl[i + hi * 8][lo] = c[i];
    // online softmax (row-owners: lanes 0..15)
    if (lane < 16) {
      int r = lane;
      float mrow = Sl[r][0];
      #pragma unroll
      for (int j = 1; j < 16; ++j) mrow = fmaxf(mrow, Sl[r][j]);
      float mnew = fmaxf(m_i, mrow);
      float alpha = __expf(m_i - mnew);
      float lsum = 0.f;
      #pragma unroll
      for (int j = 0; j < 16; ++j) {
        float p = __expf(Sl[r][j] - mnew);
        Sl[r][j] = p;
        lsum += p;
      }
      l_i = l_i * alpha + lsum;
      m_i = mnew;
      salpha[r] = alpha;
    }
    // rescale accumulators (C layout row = i + 8*hi)
    #pragma unroll
    for (int i = 0; i < 8; ++i) {
      float f = salpha[i + hi * 8];
      #pragma unroll
      for (int n = 0; n < 4; ++n) O[n][i] *= f;
    }
    // O += P V   (A = P 16x16, B = Vs 16x64 in 4 N-tiles)
    for (int kk = 0; kk < 4; ++kk) {
      int kb = (kk << 2) + hi * 2;
      v2f a;
      a.x = Sl[lo][kb]; a.y = Sl[lo][kb + 1];
      #pragma unroll
      for (int n = 0; n < 4; ++n) {
        v2f b;
        b.x = Vs[kb][n * 16 + lo];
        b.y = Vs[kb + 1][n * 16 + lo];
        O[n] = wmma4(a, b, O[n]);
      }
    }
  }
  if (lane < 16) slsum[lane] = l_i;
  #pragma unroll
  for (int i = 0; i < 8; ++i) {
    int r = i + hi * 8;
    float inv = 1.0f / slsum[r];
    size_t base = ((size_t)(s0 + r) * L_ + l) * D_;
    #pragma unroll
    for (int n = 0; n < 4; ++n) {
      int col = n * 16 + lo;
      if (col < D_) ctx[base + col] = O[n][i] * inv;
    }
  }
}

// ---------------------------------------------------------------------------
// K6a: GGNN adjacency aggregation: ag_in = A[:, :, :20] @ h, ag_out = A[:, :, 20:] @ h
// ---------------------------------------------------------------------------
__global__ __launch_bounds__(256) void aggregate_kernel(
    const float* __restrict__ A, const float* __restrict__ h,
    float* __restrict__ ag_in, float* __restrict__ ag_out) {
  int b = blockIdx.x;
  __shared__ float Hs[L_][D_];
  __shared__ float As[L_][2 * L_];
  int tid = threadIdx.x;
  for (int i = tid; i < L_ * D_; i += 256)
    Hs[i / D_][i % D_] = h[(size_t)b * (L_ * D_) + i];
  for (int i = tid; i < L_ * 2 * L_; i += 256)
    As[i / (2 * L_)][i % (2 * L_)] = A[(size_t)b * (L_ * 2 * L_) + i];
  __syncthreads();
  for (int i = tid; i < 2 * L_ * D_; i += 256) {
    int half = i / (L_ * D_);
    int j = i - half * (L_ * D_);
    int t = j / D_, d = j - t * D_;
    float s = 0.f;
    #pragma unroll
    for (int u = 0; u < L_; ++u) s += As[t][half * L_ + u] * Hs[u][d];
    float* dst = half ? ag_out : ag_in;
    dst[((size_t)b * L_ + t) * D_ + d] = s;
  }
}

// ---------------------------------------------------------------------------
// K6b1: reset = sigmoid(r_old + r_in); rh = reset * h   (note: (reset*h)@W later)
// ---------------------------------------------------------------------------
__global__ __launch_bounds__(256) void gru_reset_kernel(
    const float* __restrict__ rzh, const float* __restrict__ rzold,
    const float* __restrict__ h, float* __restrict__ rh) {
  int i = blockIdx.x * 256 + threadIdx.x;
  if (i >= BL_ * D_) return;
  int r = i / D_, e = i - r * D_;
  float x = rzold[(size_t)r * 100 + e] + rzh[(size_t)r * 150 + e];
  float reset = 1.0f / (1.0f + __expf(-x));
  rh[i] = reset * h[i];
}

// ---------------------------------------------------------------------------
// K6b2: update gate + output blend
// ---------------------------------------------------------------------------
__global__ __launch_bounds__(256) void gru_out_kernel(
    const float* __restrict__ rzh, const float* __restrict__ rzold,
    const float* __restrict__ h, const float* __restrict__ hw,
    const float* __restrict__ b_h_old, float* __restrict__ out) {
  int i = blockIdx.x * 256 + threadIdx.x;
  if (i >= BL_ * D_) return;
  int r = i / D_, e = i - r * D_;
  float z = rzold[(size_t)r * 100 + 50 + e] + rzh[(size_t)r * 150 + 50 + e];
  float update = 1.0f / (1.0f + __expf(-z));
  float hnew = tanhf(rzh[(size_t)r * 150 + 100 + e] + hw[i] + b_h_old[e]);
  out[i] = (1.0f - update) * h[i] + update * hnew;
}

// ---------------------------------------------------------------------------
// K7a: GAT context: softmax over 11 (10 neigh + self) dots, weighted sum.
// One wave per node m; butterfly reductions across 32 lanes.
// ---------------------------------------------------------------------------
__global__ __launch_bounds__(32) void gat_ctx_kernel(
    const float* __restrict__ self_vecs, const float* __restrict__ neigh,
    float* __restrict__ gctx) {
  int m = blockIdx.x;
  int lane = threadIdx.x;
  int d0 = lane, d1 = lane + 32;
  float sv0 = self_vecs[(size_t)m * D_ + d0];
  float sv1 = (d1 < D_) ? self_vecs[(size_t)m * D_ + d1] : 0.f;
  float dots[11];
  #pragma unroll
  for (int s = 0; s < 11; ++s) {
    float n0, n1;
    if (s < 10) {
      n0 = neigh[((size_t)m * 10 + s) * D_ + d0];
      n1 = (d1 < D_) ? neigh[((size_t)m * 10 + s) * D_ + d1] : 0.f;
    } else {
      n0 = sv0; n1 = sv1;
    }
    float p = sv0 * n0 + sv1 * n1;
    p += __shfl_xor(p, 16, 32);
    p += __shfl_xor(p, 8, 32);
    p += __shfl_xor(p, 4, 32);
    p += __shfl_xor(p, 2, 32);
    p += __shfl_xor(p, 1, 32);
    dots[s] = p;
  }
  float mx = dots[0];
  #pragma unroll
  for (int s = 1; s < 11; ++s) mx = fmaxf(mx, dots[s]);
  float sum = 0.f;
  float w[11];
  #pragma unroll
  for (int s = 0; s < 11; ++s) { w[s] = __expf(dots[s] - mx); sum += w[s]; }
  float inv = 1.0f / sum;
  float c0 = 0.f, c1 = 0.f;
  #pragma unroll
  for (int s = 0; s < 11; ++s) {
    float n0, n1;
    if (s < 10) {
      n0 = neigh[((size_t)m * 10 + s) * D_ + d0];
      n1 = (d1 < D_) ? neigh[((size_t)m * 10 + s) * D_ + d1] : 0.f;
    } else {
      n0 = sv0; n1 = sv1;
    }
    c0 += w[s] * inv * n0;
    c1 += w[s] * inv * n1;
  }
  gctx[(size_t)m * D_ + d0] = c0;
  if (d1 < D_) gctx[(size_t)m * D_ + d1] = c1;
}

// ---------------------------------------------------------------------------
extern "C" void kernel_launch(void* const* d_in, const int* in_sizes, int n_in,
                              void* d_out, int out_size, void* d_ws,
                              size_t ws_size, hipStream_t stream) {
  const float* item_emb   = (const float*)d_in[0];
  const float* pop_emb    = (const float*)d_in[1];
  const float* beta       = (const float*)d_in[2];
  const float* ln_w       = (const float*)d_in[3];
  const float* ln_b       = (const float*)d_in[4];
  const float* in_proj_w  = (const float*)d_in[5];
  const float* in_proj_b  = (const float*)d_in[6];
  const float* out_proj_w = (const float*)d_in[7];
  const float* out_proj_b = (const float*)d_in[8];
  const float* w_in       = (const float*)d_in[9];
  const float* b_in       = (const float*)d_in[10];
  const float* w_out      = (const float*)d_in[11];
  const float* b_out      = (const float*)d_in[12];
  const float* w_rzh      = (const float*)d_in[13];
  const float* b_rzh      = (const float*)d_in[14];
  const float* w_rz_old   = (const float*)d_in[15];
  const float* b_rz_old   = (const float*)d_in[16];
  const float* w_h_old    = (const float*)d_in[17];
  const float* b_h_old    = (const float*)d_in[18];
  const float* gat_w      = (const float*)d_in[19];
  const float* gat_b      = (const float*)d_in[20];
  const float* A          = (const float*)d_in[21];
  const float* self_vecs  = (const float*)d_in[22];
  const float* neigh_vecs = (const float*)d_in[23];
  const int* input_session = (const int*)d_in[24];
  const int* input_pop     = (const int*)d_in[25];
  const int* node_items    = (const int*)d_in[26];

  float* out    = (float*)d_out;
  float* hidden = out;                       // 1,024,000
  float* ggnn   = out + 1024000;             // 1,024,000
  float* gat    = out + 2048000;             //   256,000

  // Workspace (floats), ~50.2 MB total; regions reused via stream ordering.
  float* ws    = (float*)d_ws;
  float* emb   = ws;              // 1,024,000
  float* h     = ws + 1024000;    // 1,024,000
  float* qkv   = ws + 2048000;    // 3,072,000 ; reused as rzh
  float* ctx   = ws + 5120000;    // 1,024,000 ; reused as hw
  float* agin  = ws + 6144000;    // 1,024,000 ; reused as rh
  float* agout = ws + 7168000;    // 1,024,000
  float* hcat  = ws + 8192000;    // 2,048,000
  float* rzold = ws + 10240000;   // 2,048,000
  float* gctx  = ws + 12288000;   //   256,000
  float* rzh = qkv;
  float* hw  = ctx;
  float* rh  = agin;

  const float one_m_alpha = 1.0f - 0.9f;

  // embeddings + node gather
  gather_kernel<<<(BL_ * D_ + 255) / 256, 256, 0, stream>>>(
      item_emb, pop_emb, input_session, input_pop, node_items, emb, h);
  // DSP branch -> hidden = ALPHA * dsp
  dsp_kernel<<<B_, 256, 0, stream>>>(emb, beta, ln_w, ln_b, hidden);
  // qkv = emb @ in_proj_w^T + b   (M=20480, K=50, N=150)
  gemm_wmma<<<BL_ / 64, 128, 0, stream>>>(emb, 50, in_proj_w, in_proj_b, qkv,
                                          150, 50, 150, 0, 0.f);
  // batch attention -> ctx  (20 heads of 1024x1024, d=50)
  attn_kernel<<<dim3(B_ / 16, L_), 32, 0, stream>>>(qkv, ctx);
  // hidden += (1-ALPHA) * (ctx @ out_proj_w^T + b)
  gemm_wmma<<<BL_ / 64, 128, 0, stream>>>(ctx, 50, out_proj_w, out_proj_b,
                                          hidden, 50, 50, 50, 1, one_m_alpha);
  // GGNN
  aggregate_kernel<<<B_, 256, 0, stream>>>(A, h, agin, agout);
  gemm_wmma<<<BL_ / 64, 128, 0, stream>>>(agin, 50, w_in, b_in, hcat, 100, 50,
                                          50, 0, 0.f);
  gemm_wmma<<<BL_ / 64, 128, 0, stream>>>(agout, 50, w_out, b_out, hcat + 50,
                                          100, 50, 50, 0, 0.f);
  gemm_wmma<<<BL_ / 64, 128, 0, stream>>>(hcat, 100, w_rzh, b_rzh, rzh, 150,
                                          100, 150, 0, 0.f);
  gemm_wmma<<<BL_ / 64, 128, 0, stream>>>(h, 50, w_rz_old, b_rz_old, rzold,
                                          100, 50, 100, 0, 0.f);
  gru_reset_kernel<<<(BL_ * D_ + 255) / 256, 256, 0, stream>>>(rzh, rzold, h,
                                                               rh);
  gemm_wmma<<<BL_ / 64, 128, 0, stream>>>(rh, 50, w_h_old, nullptr, hw, 50, 50,
                                          50, 0, 0.f);
  gru_out_kernel<<<(BL_ * D_ + 255) / 256, 256, 0, stream>>>(rzh, rzold, h, hw,
                                                             b_h_old, ggnn);
  // GAT
  gat_ctx_kernel<<<M_, 32, 0, stream>>>(self_vecs, neigh_vecs, gctx);
  gemm_wmma<<<M_ / 64, 128, 0, stream>>>(gctx, 50, gat_w, gat_b, gat, 50, 50,
                                         50, 2, 0.f);
}